// AudioVisualInteractionGraph_37142877176065
// MI455X (gfx1250) — compile-verified
//
#include <hip/hip_runtime.h>
#include <stdint.h>

typedef __attribute__((ext_vector_type(2))) float v2f;
typedef __attribute__((ext_vector_type(8))) float v8f;

constexpr int Bn  = 8;     // batch
constexpr int T   = 2048;  // Tv == Ta
constexpr int D   = 256;   // feature dim
constexpr int KNN = 8;     // num_neighbors (fixed by setup_inputs)

__device__ inline v8f wmma_f32(v2f a, v2f b, v8f c) {
  return __builtin_amdgcn_wmma_f32_16x16x4_f32(false, a, false, b, (short)0, c,
                                               false, false);
}

// --------------------------------------------------------------------------
// Kernel 1: Y[b] = X[b] @ W   (per-batch GEMM, fp32 WMMA 16x16x4)
// One wave computes a 16x32 output strip: two N-tiles share the A-fragment
// (halves the strided A loads, two independent WMMA accumulator chains).
// --------------------------------------------------------------------------
__global__ void __launch_bounds__(32)
gemm_xw(const float* __restrict__ X, const float* __restrict__ W,
        float* __restrict__ Y) {
  const int m0   = blockIdx.x * 16;
  const int n0   = blockIdx.y * 32;
  const int b    = blockIdx.z;
  const int lane = threadIdx.x;
  const int lid  = lane & 15;
  const int half = lane >> 4;
  const int koff = half * 2;

  const float* Xb = X + (size_t)b * T * D;
  float*       Yb = Y + (size_t)b * T * D;

  const float* arow  = Xb + (size_t)(m0 + lid) * D + koff;
  const float* bcol0 = W + (size_t)koff * D + (n0 + lid);
  const float* bcol1 = bcol0 + 16;

  v8f c0 = {}, c1 = {};
  for (int k0 = 0; k0 < D; k0 += 4) {
    v2f a, b0, b1;
    a.x  = arow[k0];
    a.y  = arow[k0 + 1];
    b0.x = bcol0[(size_t)k0 * D];
    b0.y = bcol0[(size_t)(k0 + 1) * D];
    b1.x = bcol1[(size_t)k0 * D];
    b1.y = bcol1[(size_t)(k0 + 1) * D];
    c0 = wmma_f32(a, b0, c0);
    c1 = wmma_f32(a, b1, c1);
  }

  const int mbase = m0 + half * 8;
#pragma unroll
  for (int r = 0; r < 8; ++r) {
    Yb[(size_t)(mbase + r) * D + (n0 + lid)]      = c0[r];
    Yb[(size_t)(mbase + r) * D + (n0 + 16 + lid)] = c1[r];
  }
}

// --------------------------------------------------------------------------
// Kernel 2: squared L2 norm of each row of Y  (one wave per row)
// --------------------------------------------------------------------------
__global__ void __launch_bounds__(256)
row_norm2(const float* __restrict__ Y, float* __restrict__ nrm) {
  const int wave = threadIdx.x >> 5;
  const int lane = threadIdx.x & 31;
  const int row  = blockIdx.x * 8 + wave;   // in [0, Bn*T)
  const float* p = Y + (size_t)row * D;
  float s = 0.f;
#pragma unroll
  for (int i = 0; i < D / 32; ++i) {
    float v = p[lane + i * 32];
    s += v * v;
  }
#pragma unroll
  for (int off = 16; off > 0; off >>= 1) s += __shfl_xor(s, off, 32);
  if (lane == 0) nrm[row] = s;
}

// --------------------------------------------------------------------------
// Kernel 3: fused pairwise-dot (WMMA) + streaming top-8 per audio column.
// Block = 128 threads (4 waves) handles one (batch, 16-audio-column) tile.
// score(t,s) = |vm_t|^2 - 2 vm_t . am_s   (rank-equivalent to -graph).
// Audio tile staged in LDS with GLOBAL_LOAD_ASYNC_TO_LDS_B128 (ASYNCcnt).
// Each wave register-tiles 2 visual M-tiles per pass sharing the B-frag.
// --------------------------------------------------------------------------
__global__ void __launch_bounds__(128)
topk_kernel(const float* __restrict__ vm, const float* __restrict__ am,
            const float* __restrict__ vnorm, int* __restrict__ knn) {
  const int tilesPerB = T / 16;
  const int s0   = (blockIdx.x % tilesPerB) * 16;
  const int b    = blockIdx.x / tilesPerB;
  const int tid  = threadIdx.x;
  const int wave = tid >> 5;      // 0..3
  const int lane = tid & 31;
  const int lid  = lane & 15;
  const int half = lane >> 4;
  const int koff = half * 2;

  __shared__ float amL[16 * 260];            // padded stride -> no bank conflicts
  __shared__ float scoreL[16][8][KNN];       // [col][wave*2+half][j]
  __shared__ int   idxL[16][8][KNN];

  // ---- Stage the 16 x 256 audio tile into LDS via async memory->LDS DMA.
  // 1024 16B chunks, 8 per thread. Row r occupies amL[r*260 .. r*260+255].
  {
    const float* amB = am + ((size_t)b * T + s0) * D;
#pragma unroll
    for (int n = 0; n < 8; ++n) {
      const int i    = tid + n * 128;    // float4 chunk index
      const int row  = i >> 6;
      const int col4 = (i & 63) * 4;
      const uint32_t ldsOff =
          (uint32_t)(uintptr_t)(const void*)&amL[row * 260 + col4];
      const float* gp = amB + (size_t)row * D + col4;
      asm volatile("global_load_async_to_lds_b128 %0, %1, off"
                   :: "v"(ldsOff), "v"(gp)
                   : "memory");
    }
    asm volatile("s_wait_asynccnt 0x0" ::: "memory");
  }
  __syncthreads();

  float best[KNN];
  int   bidx[KNN];
#pragma unroll
  for (int j = 0; j < KNN; ++j) { best[j] = 3.0e38f; bidx[j] = 0x7fffffff; }

  const float* vmB  = vm + (size_t)b * T * D;
  const float* vnB  = vnorm + (size_t)b * T;
  const float* brow = amL + lid * 260 + koff;

  // Register top-8 insertion for one 16x16 C tile (8 rows per lane).
  auto consider = [&](const v8f& c, int mbase) {
#pragma unroll
    for (int r = 0; r < 8; ++r) {
      const int   m     = mbase + r;
      const float score = vnB[m] - 2.0f * c[r];
      // strict < is the exact gate: t increases monotonically, so a candidate
      // equal to best[7] has a larger index and must be excluded (tie->low idx)
      if (score < best[KNN - 1]) {
        float cs = score; int ci = m;
#pragma unroll
        for (int p = 0; p < KNN; ++p) {
          const bool  sw = (cs < best[p]) || (cs == best[p] && ci < bidx[p]);
          const float tb = best[p]; const int ti = bidx[p];
          best[p] = sw ? cs : tb;  bidx[p] = sw ? ci : ti;
          cs      = sw ? tb : cs;  ci      = sw ? ti : ci;
        }
      }
    }
  };

  for (int pt = wave; pt < tilesPerB / 2; pt += 4) {
    const int m0 = pt * 32;                       // two 16-row tiles
    const float* arow0 = vmB + (size_t)(m0 + lid) * D + koff;
    const float* arow1 = arow0 + (size_t)16 * D;

    v8f c0 = {}, c1 = {};
    for (int k0 = 0; k0 < D; k0 += 4) {
      v2f bf, a0, a1;
      bf.x = brow[k0];
      bf.y = brow[k0 + 1];
      a0.x = arow0[k0];
      a0.y = arow0[k0 + 1];
      a1.x = arow1[k0];
      a1.y = arow1[k0 + 1];
      c0 = wmma_f32(a0, bf, c0);   // two independent accumulator chains
      c1 = wmma_f32(a1, bf, c1);
    }
    consider(c0, m0 + half * 8);        // rows processed in increasing m
    consider(c1, m0 + 16 + half * 8);
  }

  // Publish the 8 sorted partial lists per column (2 per wave).
#pragma unroll
  for (int j = 0; j < KNN; ++j) {
    scoreL[lid][wave * 2 + half][j] = best[j];
    idxL[lid][wave * 2 + half][j]   = bidx[j];
  }
  __syncthreads();

  // 8-way merge of sorted lists; threads 0..15 each own one audio column.
  if (tid < 16) {
    int head[8] = {0, 0, 0, 0, 0, 0, 0, 0};
    int* out = knn + (size_t)b * KNN * T;   // knn[b][j][s]
    for (int j = 0; j < KNN; ++j) {
      float bs = 3.9e38f; int bi = 0x7fffffff; int bsrc = 0;
#pragma unroll
      for (int l = 0; l < 8; ++l) {
        const int   h  = head[l];           // h <= j <= 7, always in range
        const float v  = scoreL[tid][l][h];
        const int   ix = idxL[tid][l][h];
        if (v < bs || (v == bs && ix < bi)) { bs = v; bi = ix; bsrc = l; }
      }
      head[bsrc]++;
      out[(size_t)j * T + s0 + tid] = bi;
    }
  }
}

// --------------------------------------------------------------------------
// Kernel 4: gather top-k rows of raw V and A, mean over the Ta axis.
// One block per (b, j); one thread per feature dim.
// --------------------------------------------------------------------------
__global__ void __launch_bounds__(256)
mean_gather(const float* __restrict__ V, const float* __restrict__ A,
            const int* __restrict__ knn, float* __restrict__ out) {
  const int j = blockIdx.x;   // neighbor slot
  const int b = blockIdx.y;
  const int d = threadIdx.x;  // 0..255

  __shared__ int idxL[T];
  const int* src = knn + ((size_t)b * KNN + j) * T;
  for (int i = threadIdx.x; i < T; i += 256) idxL[i] = src[i];
  __syncthreads();

  const float* Vb = V + (size_t)b * T * D;
  const float* Ab = A + (size_t)b * T * D;
  float sv = 0.f, sa = 0.f;
  for (int s = 0; s < T; ++s) {
    const int t = idxL[s];
    sv += Vb[(size_t)t * D + d];
    sa += Ab[(size_t)t * D + d];
  }
  const float inv = 1.0f / (float)T;
  float* outv = out;                            // [B, k, D]
  float* outa = out + (size_t)Bn * KNN * D;     // [B, k, D]
  outv[((size_t)b * KNN + j) * D + d] = sv * inv;
  outa[((size_t)b * KNN + j) * D + d] = sa * inv;
}

// --------------------------------------------------------------------------
extern "C" void kernel_launch(void* const* d_in, const int* in_sizes, int n_in,
                              void* d_out, int out_size, void* d_ws,
                              size_t ws_size, hipStream_t stream) {
  (void)in_sizes; (void)n_in; (void)out_size; (void)ws_size;

  const float* V  = (const float*)d_in[0];  // [B,T,D]
  const float* A  = (const float*)d_in[1];  // [B,T,D]
  const float* Wv = (const float*)d_in[2];  // [D,D]
  const float* Wa = (const float*)d_in[3];  // [D,D]
  float* out = (float*)d_out;               // [B,k,D] ++ [B,k,D]

  // Workspace layout (~34 MB): vm, am, vnorm, knn indices.
  float* vm    = (float*)d_ws;
  float* am    = vm + (size_t)Bn * T * D;
  float* vnorm = am + (size_t)Bn * T * D;
  int*   knn   = (int*)(vnorm + (size_t)Bn * T);

  dim3 ggrid(T / 16, D / 32, Bn);
  gemm_xw<<<ggrid, 32, 0, stream>>>(V, Wv, vm);
  gemm_xw<<<ggrid, 32, 0, stream>>>(A, Wa, am);
  row_norm2<<<(Bn * T) / 8, 256, 0, stream>>>(vm, vnorm);
  topk_kernel<<<Bn * (T / 16), 128, 0, stream>>>(vm, am, vnorm, knn);
  mean_gather<<<dim3(KNN, Bn), 256, 0, stream>>>(V, A, knn, out);
}